// StochasticFullRGNLoss_83056077570643
// MI455X (gfx1250) — compile-verified
//
#include <hip/hip_runtime.h>
#include <hip/hip_bf16.h>

#define N_GROUPS 512
#define GROUP_LEN 256
#define SAMPLES 128

typedef __attribute__((ext_vector_type(2))) float v2f;
typedef __attribute__((ext_vector_type(8))) float v8f;

// Upper-triangular 16x16-tile index map: t -> (i<<4)|j, i<=j, 8x8 tile grid.
__device__ const unsigned char kTileIJ[36] = {
    0x00,0x01,0x02,0x03,0x04,0x05,0x06,0x07,
    0x11,0x12,0x13,0x14,0x15,0x16,0x17,
    0x22,0x23,0x24,0x25,0x26,0x27,
    0x33,0x34,0x35,0x36,0x37,
    0x44,0x45,0x46,0x47,
    0x55,0x56,0x57,
    0x66,0x67,
    0x77
};

// One workgroup (256 threads = 8 wave32) per protein group.
// Pairwise distances via symmetric Gram matrices computed with
// V_WMMA_F32_16X16X4_F32 (K=3 padded to 4 with a zero column):
//   d^2(m,n) = ||p_m||^2 + ||p_n||^2 - 2 * (P P^T)(m,n)
// Only the 36 upper-triangular tiles of the 8x8 tile grid are computed;
// off-diagonal elements are weighted by 2 (the matrix is symmetric and the
// diagonal is exactly zero).
__launch_bounds__(256)
__global__ void StochasticFullRGNLoss_83056077570643_kernel(
    const float* __restrict__ inputs,    // [N,3,3] flat
    const float* __restrict__ target,    // [N,3,3] flat
    const int*   __restrict__ positions, // [G,S] flat
    float* __restrict__ out)             // scalar
{
    __shared__ float xs[SAMPLES][4];   // padded K=4, xs[s][3]=0
    __shared__ float ts[SAMPLES][4];
    __shared__ float nx[SAMPLES];      // squared norms (inputs)
    __shared__ float ntg[SAMPLES];     // squared norms (target)
    __shared__ float wsum[8];

    const int g   = blockIdx.x;
    const int tid = threadIdx.x;

    // ---- Phase 1: gather CA atom (row 1 of the 3x3 frame) into LDS ----
    if (tid < SAMPLES) {
        const int s   = tid;
        const int idx = positions[g * SAMPLES + s];
        const float* p = inputs + (size_t)idx * 9 + 3;   // row 1 = elems 3..5
        const float a = p[0], b = p[1], c = p[2];
        xs[s][0] = a; xs[s][1] = b; xs[s][2] = c; xs[s][3] = 0.0f;
        nx[s] = a * a + b * b + c * c;
    } else if (tid < 2 * SAMPLES) {
        const int s   = tid - SAMPLES;
        const int idx = positions[g * SAMPLES + s];
        const float* p = target + (size_t)idx * 9 + 3;
        const float a = p[0], b = p[1], c = p[2];
        ts[s][0] = a; ts[s][1] = b; ts[s][2] = c; ts[s][3] = 0.0f;
        ntg[s] = a * a + b * b + c * c;
    }
    __syncthreads();

    const int wave = tid >> 5;
    const int lane = tid & 31;
    const int half = lane >> 4;      // 0: lanes 0-15, 1: lanes 16-31
    const int l16  = lane & 15;
    const int k0   = half * 2;       // A/B layout: lane-half selects K pair

    float acc = 0.0f;

    // ---- Phase 2: round-robin over the 36 upper-triangular tiles ----
    #pragma unroll
    for (int k = 0; k < 5; ++k) {
        const int t = wave + 8 * k;      // wave-uniform -> EXEC stays full
        if (t < 36) {
            const int ij = (int)kTileIJ[t];
            const int ti = ij >> 4;      // tile row (M block)
            const int tj = ij & 15;      // tile col (N block), tj >= ti

            const int arow = ti * 16 + l16;
            const int bcol = tj * 16 + l16;

            v2f ax; ax.x = xs[arow][k0]; ax.y = xs[arow][k0 + 1];
            v2f av; av.x = ts[arow][k0]; av.y = ts[arow][k0 + 1];
            v2f bx; bx.x = xs[bcol][k0]; bx.y = xs[bcol][k0 + 1];
            v2f bv; bv.x = ts[bcol][k0]; bv.y = ts[bcol][k0 + 1];

            v8f cx = {};
            v8f ct = {};
            // 8 args: (neg_a, A, neg_b, B, c_mod, C, reuse_a, reuse_b)
            cx = __builtin_amdgcn_wmma_f32_16x16x4_f32(
                     false, ax, false, bx, (short)0, cx, false, false);
            ct = __builtin_amdgcn_wmma_f32_16x16x4_f32(
                     false, av, false, bv, (short)0, ct, false, false);

            const int n    = tj * 16 + l16;
            const float nxn = nx[n];
            const float ntn = ntg[n];
            const bool  diag = (ti == tj);

            #pragma unroll
            for (int r = 0; r < 8; ++r) {
                // C/D layout: VGPR r -> M=r (lanes 0-15) / M=r+8 (lanes 16-31)
                const int m = ti * 16 + r + 8 * half;
                const float d2x = fmaxf(nx[m]  + nxn - 2.0f * cx[r], 0.0f);
                const float d2t = fmaxf(ntg[m] + ntn - 2.0f * ct[r], 0.0f);
                // raw v_sqrt_f32 (~1 ulp) -- avoids the IEEE fixup sequence;
                // diagonal elements are exactly 0 and get weight 0 anyway.
                const float dx = __builtin_amdgcn_sqrtf(d2x);
                const float dt = __builtin_amdgcn_sqrtf(d2t);
                const float df = dx - dt;
                // symmetry weight: off-diag tiles count twice; diagonal tile
                // counts its strict upper triangle twice, diagonal is zero.
                const float w = diag ? ((n > m) ? 2.0f : 0.0f) : 2.0f;
                acc = fmaf(w * df, df, acc);
            }
        }
    }

    // ---- Phase 3: reduction (wave32 shuffle, then LDS, then one atomic) ----
    #pragma unroll
    for (int off = 16; off >= 1; off >>= 1)
        acc += __shfl_xor(acc, off, 32);
    if (lane == 0) wsum[wave] = acc;
    __syncthreads();

    if (tid == 0) {
        float total = 0.0f;
        #pragma unroll
        for (int w = 0; w < 8; ++w) total += wsum[w];
        const float scale =
            1.0f / ((float)N_GROUPS * (float)SAMPLES * (float)SAMPLES);
        atomicAdd(out, total * scale);
    }
}

extern "C" void kernel_launch(void* const* d_in, const int* in_sizes, int n_in,
                              void* d_out, int out_size, void* d_ws, size_t ws_size,
                              hipStream_t stream) {
    (void)in_sizes; (void)n_in; (void)d_ws; (void)ws_size;
    const float* inputs    = (const float*)d_in[0];
    const float* target    = (const float*)d_in[1];
    const int*   positions = (const int*)d_in[2];
    float*       out       = (float*)d_out;

    // Zero the scalar accumulator (harness poisons d_out; async memset is
    // graph-capture safe).
    hipMemsetAsync(out, 0, (size_t)out_size * sizeof(float), stream);

    StochasticFullRGNLoss_83056077570643_kernel<<<N_GROUPS, 256, 0, stream>>>(
        inputs, target, positions, out);
}